// RetinaNet_82463372083573
// MI455X (gfx1250) — compile-verified
//
#include <hip/hip_runtime.h>
#include <hip/hip_bf16.h>
#include <stdint.h>

#define A_N      76725
#define C_N      80
#define APAD     76800          // A padded to 1024*75 (and to 16B chunks)
#define TPB      1024           // 32 wave32 waves
#define KW       2400           // keep-bitmask words (>= ceil(A/32))
#define MAX_DET  200
#define IMG_F    640.0f
#define SCORE_THR 0.1f

// transpose tiling
#define TA       128            // anchors per transpose tile (APAD/TA = 600 blocks)
#define CPADT    84             // padded row stride (84*4B = 336B = 21*16B; 84%64=20)

// ---------------------------------------------------------------------------
// Kernel 1: decode + clip boxes (anchors xyxy + deltas -> clipped xyxy)
// ---------------------------------------------------------------------------
__global__ __launch_bounds__(256) void decode_boxes(const float* __restrict__ anchors,
                                                    const float* __restrict__ reg,
                                                    float* __restrict__ boxes) {
    int j = blockIdx.x * 256 + threadIdx.x;
    if (j >= A_N) return;
    float4 a = reinterpret_cast<const float4*>(anchors)[j];
    float4 r = reinterpret_cast<const float4*>(reg)[j];
    float aw = a.z - a.x, ah = a.w - a.y;
    float ax = a.x + 0.5f * aw, ay = a.y + 0.5f * ah;
    float cx = ax + r.x * 0.1f * aw;
    float cy = ay + r.y * 0.1f * ah;
    float w  = aw * expf(r.z * 0.2f);
    float h  = ah * expf(r.w * 0.2f);
    float4 b;
    b.x = fminf(fmaxf(cx - 0.5f * w, 0.0f), IMG_F);
    b.y = fminf(fmaxf(cy - 0.5f * h, 0.0f), IMG_F);
    b.z = fminf(fmaxf(cx + 0.5f * w, 0.0f), IMG_F);
    b.w = fminf(fmaxf(cy + 0.5f * h, 0.0f), IMG_F);
    reinterpret_cast<float4*>(boxes)[j] = b;
}

// ---------------------------------------------------------------------------
// Kernel 2: LDS-tiled transpose [A,C] -> [C,APAD] with score-threshold mask.
// Phase 1: async-DMA the tile (contiguous in [A,C] layout) into LDS in 16B
// chunks (ASYNCcnt path). Phase 2: mask on LDS read, fully coalesced writes.
// ---------------------------------------------------------------------------
__global__ __launch_bounds__(TPB) void mask_transpose(const float* __restrict__ cls,
                                                      float* __restrict__ scoresT) {
    __shared__ float tile[TA * CPADT];          // 43,008 B
    const int tid = threadIdx.x;
    const int j0  = blockIdx.x * TA;            // first anchor of this tile
    const int nrows = (j0 + TA <= A_N) ? TA : (A_N - j0);   // valid anchors in tile

    // Phase 1: DMA rows (80 floats = 20 x b128 chunks each) into padded LDS rows
    const float* gsrc = cls + (size_t)j0 * C_N; // tile is contiguous in memory
    for (int i = tid; i < TA * (C_N / 4); i += TPB) {       // 2560 chunks
        int row = i / (C_N / 4);
        int k   = i % (C_N / 4);
        if (row < nrows) {
            unsigned           lds = (unsigned)(uintptr_t)(&tile[row * CPADT + k * 4]);
            unsigned long long ga  = (unsigned long long)(uintptr_t)(gsrc + row * C_N + k * 4);
            asm volatile("global_load_async_to_lds_b128 %0, %1, off"
                         :: "v"(lds), "v"(ga) : "memory");
        }
    }
    asm volatile("s_wait_asynccnt 0" ::: "memory");
    __syncthreads();

    // Phase 2: masked transpose out; consecutive threads -> consecutive anchors
    for (int o = tid; o < C_N * TA; o += TPB) {             // 10240 elements
        int c  = o >> 7;                                     // o / TA
        int jj = o & (TA - 1);                               // o % TA
        float v = -1.0f;
        if (jj < nrows) {
            float raw = tile[jj * CPADT + c];
            v = (raw > SCORE_THR) ? raw : -1.0f;
        }
        scoresT[(size_t)c * APAD + j0 + jj] = v;
    }
}

// ---------------------------------------------------------------------------
// Kernel 3: one workgroup per class; whole score vector lives in LDS (300KB,
// a CDNA5-only capability), staged by async global->LDS DMA (ASYNCcnt).
// 200 sequential argmax + IoU-suppress rounds, then emit outputs.
// ---------------------------------------------------------------------------
__global__ __launch_bounds__(TPB) void nms_per_class(const float* __restrict__ scoresT,
                                                     const float* __restrict__ boxes,
                                                     float* __restrict__ out_scores,
                                                     int*   __restrict__ out_labels,
                                                     float* __restrict__ out_boxes,
                                                     unsigned char* __restrict__ out_keep) {
    extern __shared__ float s_sc[];            // APAD floats = 300 KB dynamic LDS
    __shared__ float    rv[32];
    __shared__ int      ri[32];
    __shared__ float    sbx, sby, sbz, sbw, sarea;
    __shared__ int      svalid;
    __shared__ unsigned keepw[KW];             // per-anchor keep bitmask

    const int tid = threadIdx.x;
    const int cls = blockIdx.x;
    const float*  gsc = scoresT + (size_t)cls * APAD;
    const float4* bx  = reinterpret_cast<const float4*>(boxes);

    for (int w = tid; w < KW; w += TPB) keepw[w] = 0u;

    // --- async-DMA stage masked scores into LDS (gfx1250 path, ASYNCcnt) ---
    for (int i = tid; i < APAD / 4; i += TPB) {
        unsigned           lds = (unsigned)(uintptr_t)(&s_sc[i * 4]);
        unsigned long long ga  = (unsigned long long)(uintptr_t)(gsc + i * 4);
        asm volatile("global_load_async_to_lds_b128 %0, %1, off"
                     :: "v"(lds), "v"(ga) : "memory");
    }
    asm volatile("s_wait_asynccnt 0" ::: "memory");
    __syncthreads();

    // --- greedy NMS: MAX_DET dependent rounds ---
    for (int it = 0; it < MAX_DET; ++it) {
        // block-wide argmax (first-index tie-break, matches jnp.argmax)
        float bv = -2.0f; int bi = 0;
        for (int j = tid; j < APAD; j += TPB) {          // exactly 75 LDS reads
            float v = s_sc[j];
            if (v > bv) { bv = v; bi = j; }
        }
        for (int off = 16; off > 0; off >>= 1) {         // wave32 shuffle reduce
            float ov = __shfl_down(bv, off, 32);
            int   oi = __shfl_down(bi, off, 32);
            if (ov > bv || (ov == bv && oi < bi)) { bv = ov; bi = oi; }
        }
        const int lane = tid & 31, wid = tid >> 5;
        if (lane == 0) { rv[wid] = bv; ri[wid] = bi; }
        __syncthreads();
        if (wid == 0) {                                   // 32 waves -> 32 entries
            bv = rv[lane]; bi = ri[lane];
            for (int off = 16; off > 0; off >>= 1) {
                float ov = __shfl_down(bv, off, 32);
                int   oi = __shfl_down(bi, off, 32);
                if (ov > bv || (ov == bv && oi < bi)) { bv = ov; bi = oi; }
            }
            if (lane == 0) {
                if (bv > 0.0f) {
                    svalid = 1;
                    float4 b = bx[bi];
                    sbx = b.x; sby = b.y; sbz = b.z; sbw = b.w;
                    sarea = (b.z - b.x) * (b.w - b.y);
                    keepw[bi >> 5] |= (1u << (bi & 31));
                    s_sc[bi] = -1.0f;                     // self-suppress (idx==i)
                } else {
                    svalid = 0;
                    keepw[0] &= ~1u;  // reference quirk: argmax of all -1 is 0,
                }                     // keep[0] <- False once pool exhausted
            }
        }
        __syncthreads();
        if (!svalid) break;                               // state is now fixed

        // IoU sweep: suppress overlaps (skip already-dead anchors)
        const float px1 = sbx, py1 = sby, px2 = sbz, py2 = sbw, pa = sarea;
        for (int j = tid; j < APAD; j += TPB) {
            float v = s_sc[j];
            if (v > 0.0f) {                               // pad entries are -1
                float4 q = bx[j];                         // L2-resident, coalesced
                float ix1 = fmaxf(px1, q.x);
                float iy1 = fmaxf(py1, q.y);
                float ix2 = fminf(px2, q.z);
                float iy2 = fminf(py2, q.w);
                float inter = fmaxf(ix2 - ix1, 0.0f) * fmaxf(iy2 - iy1, 0.0f);
                float qa = (q.z - q.x) * (q.w - q.y);
                // iou > 0.5  <=>  inter > 0.5*union  (no divide, same NaN/0 result)
                if (inter > 0.5f * (pa + qa - inter)) s_sc[j] = -1.0f;
            }
        }
        __syncthreads();
    }
    __syncthreads();

    // --- emit outputs for this class row ---
    const size_t rowo = (size_t)cls * A_N;
    for (int j = tid; j < A_N; j += TPB) {
        bool k = (keepw[j >> 5] >> (j & 31)) & 1u;
        float sc = 0.0f;
        if (k) sc = gsc[j];                 // masked value == original score (>0.1)
        out_scores[rowo + j] = sc;
        out_labels[rowo + j] = k ? cls : -1;
        float4 ob = make_float4(0.0f, 0.0f, 0.0f, 0.0f);
        if (k) ob = bx[j];
        reinterpret_cast<float4*>(out_boxes)[rowo + j] = ob;
        out_keep[rowo + j] = k ? 1 : 0;
    }
}

// ---------------------------------------------------------------------------
extern "C" void kernel_launch(void* const* d_in, const int* in_sizes, int n_in,
                              void* d_out, int out_size, void* d_ws, size_t ws_size,
                              hipStream_t stream) {
    const float* classification = (const float*)d_in[0];  // [1, A, C] f32
    const float* regression     = (const float*)d_in[1];  // [1, A, 4] f32
    const float* anchors        = (const float*)d_in[2];  // [A, 4]    f32

    // workspace: boxes [A*4] f32, then scoresT [C][APAD] f32 (16B aligned)
    float* boxes   = (float*)d_ws;
    float* scoresT = boxes + (size_t)A_N * 4;

    // outputs, flat in reference return order:
    // final_scores [C,A] f32 | final_labels [C,A] i32 | final_boxes [C,A,4] f32 | keep [C,A] u8
    float*         out_scores = (float*)d_out;
    int*           out_labels = (int*)(out_scores + (size_t)C_N * A_N);
    float*         out_boxes  = (float*)(out_labels + (size_t)C_N * A_N);
    unsigned char* out_keep   = (unsigned char*)(out_boxes + (size_t)C_N * A_N * 4);

    decode_boxes<<<(A_N + 255) / 256, 256, 0, stream>>>(anchors, regression, boxes);
    mask_transpose<<<APAD / TA, TPB, 0, stream>>>(classification, scoresT);
    nms_per_class<<<C_N, TPB, APAD * sizeof(float), stream>>>(
        scoresT, boxes, out_scores, out_labels, out_boxes, out_keep);
}